// WindowAttention_86723979641062
// MI455X (gfx1250) — compile-verified
//
#include <hip/hip_runtime.h>
#include <hip/hip_bf16.h>
#include <math.h>

// ---------------- problem constants ----------------
#define WSZ     7
#define NTOK    49          // tokens per window
#define NPAD    64          // padded tokens (WMMA tiles)
#define HEADS   12
#define HD      32
#define EMBED   384
#define QKVC    1152        // 3 * HEADS * HD
#define HW      56
#define BATCH   16
#define DISP    3
#define NWIN    64          // 8x8 windows
#define TOKENS  (BATCH*HW*HW)   // 50176
#define SCALE   0.17677669529663687f  // 32^-0.5

typedef float v2f __attribute__((ext_vector_type(2)));
typedef float v8f __attribute__((ext_vector_type(8)));

// D = A(16x4) * B(4x16) + C, fp32 WMMA (full precision; reference is fp32)
__device__ __forceinline__ v8f wmma4(v2f a, v2f b, v8f c) {
    return __builtin_amdgcn_wmma_f32_16x16x4_f32(
        /*neg_a=*/false, a, /*neg_b=*/false, b,
        /*c_mod=*/(short)0, c, /*reuse_a=*/false, /*reuse_b=*/false);
}

__device__ __forceinline__ v2f lds_pair(const float* p) {
    return *(const v2f*)p;      // 8B-aligned contiguous pair -> ds_load_b64
}

// ================= GEMM tiling =================
// 128 threads = 4 waves; each wave owns a 32x64 strip (2 M-tiles x 4 N-tiles)
#define BM 128
#define BN 64
#define KC 64
#define AP 68   // LDS row stride (68 % 64 == 4 -> conflict-free fragment reads)

// ================= Kernel 1: QKV projection (fused input roll) =================
// qkv[t][c] = sum_k x_rolled[t][k] * w_qkv[c][k]
__global__ __launch_bounds__(128)
void qkv_gemm(const float* __restrict__ x, const float* __restrict__ w_qkv,
              float* __restrict__ qkv)
{
    __shared__ float As[BM * AP];   // A tile: [row][k]
    __shared__ float Bt[BN * AP];   // B tile transposed: [n][k]
    __shared__ long  rowoff[BM];

    const int tid    = threadIdx.x;
    const int m_base = blockIdx.x * BM;
    const int n_base = blockIdx.y * BN;

    {   // rolled row offsets (roll -3,-3)
        int t = m_base + tid;
        int b = t / (HW*HW);
        int r = t - b*HW*HW;
        int i = r / HW, j = r - i*HW;
        int is = (i + DISP) % HW, js = (j + DISP) % HW;
        rowoff[tid] = ((long)(b*HW + is)*HW + js) * EMBED;
    }

    const int wave = tid >> 5;
    const int lane = tid & 31;
    const int lr   = lane & 15;           // M (for A) or N (for B) index
    const int kh   = (lane >> 4) << 1;    // 0 or 2 (K pair select)
    const int m0   = wave * 32;           // two 16-row tiles per wave

    v8f acc[2][4] = {};

    for (int kk = 0; kk < EMBED; kk += KC) {
        __syncthreads();
        // stage A: 128 rows x 64 k
        #pragma unroll
        for (int it = 0; it < (BM*KC)/128; ++it) {
            int idx = it*128 + tid;
            int r = idx >> 6, c = idx & 63;
            As[r*AP + c] = x[rowoff[r] + kk + c];
        }
        // stage B transposed: Bt[n][k] = w_qkv[n_base+n][kk+k]  (contiguous writes)
        #pragma unroll
        for (int it = 0; it < (KC*BN)/128; ++it) {
            int idx = it*128 + tid;
            int n = idx >> 6, k = idx & 63;
            Bt[n*AP + k] = w_qkv[(long)(n_base + n)*EMBED + kk + k];
        }
        __syncthreads();

        for (int ks = 0; ks < KC; ks += 4) {
            v2f a0 = lds_pair(As + (m0      + lr)*AP + ks + kh);
            v2f a1 = lds_pair(As + (m0 + 16 + lr)*AP + ks + kh);
            #pragma unroll
            for (int nt = 0; nt < 4; ++nt) {
                v2f b = lds_pair(Bt + (nt*16 + lr)*AP + ks + kh);
                acc[0][nt] = wmma4(a0, b, acc[0][nt]);
                acc[1][nt] = wmma4(a1, b, acc[1][nt]);
            }
        }
    }

    const int nIdx  = lane & 15;
    const int mHalf = (lane >> 4) * 8;
    for (int mt = 0; mt < 2; ++mt) {
        for (int nt = 0; nt < 4; ++nt) {
            int c = n_base + nt*16 + nIdx;
            #pragma unroll
            for (int j = 0; j < 8; ++j) {
                int m = m_base + m0 + mt*16 + mHalf + j;
                qkv[(long)m * QKVC + c] = acc[mt][nt][j];
            }
        }
    }
}

// ================= Kernel 2: windowed attention =================
// One block (4 waves) per (batch, head, window).
#define QP 36   // LDS stride for 32-wide Q/K rows (conflict-free fragment reads)
#define SP 68   // LDS stride for 64-wide S rows

__global__ __launch_bounds__(128)
void attn_kernel(const float* __restrict__ qkv, const float* __restrict__ pos_emb,
                 float* __restrict__ attnout)
{
    int bid = blockIdx.x;
    int w = bid % NWIN; int t2 = bid / NWIN;
    int h = t2 % HEADS; int b = t2 / HEADS;
    int wi = w >> 3, wj = w & 7;

    __shared__ float Qs[NPAD * QP];   // [token][d]
    __shared__ float Ks[NPAD * QP];   // [token][d]
    __shared__ float Vt[HD * SP];     // transposed: [d][token]
    __shared__ float Ss[NPAD * SP];   // scores / probabilities

    const int tid = threadIdx.x;

    // stage Q,K,V (tokens 49..63 zero-padded)
    #pragma unroll
    for (int it = 0; it < (NPAD*HD)/128; ++it) {
        int idx = it*128 + tid;
        int r = idx >> 5, d = idx & 31;
        float q = 0.f, k = 0.f, v = 0.f;
        if (r < NTOK) {
            int i = wi*WSZ + r/WSZ;
            int j = wj*WSZ + r%WSZ;
            long base = ((long)(b*HW + i)*HW + j)*QKVC + h*HD + d;
            q = qkv[base];
            k = qkv[base + EMBED];
            v = qkv[base + 2*EMBED];
        }
        Qs[r*QP + d] = q;
        Ks[r*QP + d] = k;
        Vt[d*SP + r] = v;
    }
    __syncthreads();

    const int wave = tid >> 5, lane = tid & 31;
    const int lr = lane & 15, kh = (lane >> 4) << 1;
    const int m0 = wave * 16;
    const int nIdx = lane & 15, mHalf = (lane >> 4) * 8;

    // S = Q * K^T   (B[k][n] = K[n][k], contiguous pairs in Ks rows)
    v8f s[4] = {};
    for (int ks = 0; ks < HD; ks += 4) {
        v2f a = lds_pair(Qs + (m0+lr)*QP + ks + kh);
        #pragma unroll
        for (int nt = 0; nt < 4; ++nt) {
            v2f bb = lds_pair(Ks + (nt*16+lr)*QP + ks + kh);
            s[nt] = wmma4(a, bb, s[nt]);
        }
    }
    for (int nt = 0; nt < 4; ++nt)
        #pragma unroll
        for (int j = 0; j < 8; ++j)
            Ss[(m0+mHalf+j)*SP + nt*16 + nIdx] = s[nt][j];
    __syncthreads();

    // scale + rel-pos bias + shift masks + softmax (one row per thread)
    if (tid < NPAD) {
        int r = tid;
        if (r < NTOK) {
            int ri = r/WSZ, rj = r%WSZ;
            float mx = -__builtin_inff();
            for (int c = 0; c < NTOK; ++c) {
                int ci = c/WSZ, cj = c%WSZ;
                float v = Ss[r*SP + c]*SCALE
                        + pos_emb[(ci - ri + 6)*13 + (cj - rj + 6)];
                if (wi == 7 && ((r >= 28) != (c >= 28))) v = -__builtin_inff();
                if (wj == 7 && ((rj >= 4) != (cj >= 4))) v = -__builtin_inff();
                mx = fmaxf(mx, v);
            }
            float sum = 0.f;
            for (int c = 0; c < NTOK; ++c) {
                int ci = c/WSZ, cj = c%WSZ;
                float v = Ss[r*SP + c]*SCALE
                        + pos_emb[(ci - ri + 6)*13 + (cj - rj + 6)];
                if (wi == 7 && ((r >= 28) != (c >= 28))) v = -__builtin_inff();
                if (wj == 7 && ((rj >= 4) != (cj >= 4))) v = -__builtin_inff();
                float e = expf(v - mx);
                Ss[r*SP + c] = e;
                sum += e;
            }
            float inv = 1.f / sum;
            for (int c = 0; c < NTOK; ++c) Ss[r*SP + c] *= inv;
            for (int c = NTOK; c < NPAD; ++c) Ss[r*SP + c] = 0.f;
        } else {
            for (int c = 0; c < NPAD; ++c) Ss[r*SP + c] = 0.f;
        }
    }
    __syncthreads();

    // O = P * V   (B[k][n] = V[k][n] = Vt[n][k], contiguous pairs in Vt rows)
    v8f o[2] = {};
    for (int ks = 0; ks < NPAD; ks += 4) {
        v2f a = lds_pair(Ss + (m0+lr)*SP + ks + kh);
        #pragma unroll
        for (int nt = 0; nt < 2; ++nt) {
            v2f bb = lds_pair(Vt + (nt*16+lr)*SP + ks + kh);
            o[nt] = wmma4(a, bb, o[nt]);
        }
    }
    for (int nt = 0; nt < 2; ++nt) {
        int d = h*HD + nt*16 + nIdx;
        #pragma unroll
        for (int j = 0; j < 8; ++j) {
            int r = m0 + mHalf + j;
            if (r < NTOK) {
                int i  = wi*WSZ + r/WSZ;
                int jj = wj*WSZ + r%WSZ;
                attnout[((long)(b*HW + i)*HW + jj)*EMBED + d] = o[nt][j];
            }
        }
    }
}

// ================= Kernel 3: output projection + bias + inverse roll =================
__global__ __launch_bounds__(128)
void out_gemm(const float* __restrict__ ain, const float* __restrict__ w_out,
              const float* __restrict__ b_out, float* __restrict__ out)
{
    __shared__ float As[BM * AP];
    __shared__ float Bt[BN * AP];

    const int tid    = threadIdx.x;
    const int m_base = blockIdx.x * BM;
    const int n_base = blockIdx.y * BN;

    const int wave = tid >> 5;
    const int lane = tid & 31;
    const int lr   = lane & 15;
    const int kh   = (lane >> 4) << 1;
    const int m0   = wave * 32;

    v8f acc[2][4] = {};

    for (int kk = 0; kk < EMBED; kk += KC) {
        __syncthreads();
        #pragma unroll
        for (int it = 0; it < (BM*KC)/128; ++it) {
            int idx = it*128 + tid;
            int r = idx >> 6, c = idx & 63;
            As[r*AP + c] = ain[(long)(m_base + r)*EMBED + kk + c];
        }
        #pragma unroll
        for (int it = 0; it < (KC*BN)/128; ++it) {
            int idx = it*128 + tid;
            int n = idx >> 6, k = idx & 63;
            Bt[n*AP + k] = w_out[(long)(n_base + n)*EMBED + kk + k];
        }
        __syncthreads();

        for (int ks = 0; ks < KC; ks += 4) {
            v2f a0 = lds_pair(As + (m0      + lr)*AP + ks + kh);
            v2f a1 = lds_pair(As + (m0 + 16 + lr)*AP + ks + kh);
            #pragma unroll
            for (int nt = 0; nt < 4; ++nt) {
                v2f b = lds_pair(Bt + (nt*16 + lr)*AP + ks + kh);
                acc[0][nt] = wmma4(a0, b, acc[0][nt]);
                acc[1][nt] = wmma4(a1, b, acc[1][nt]);
            }
        }
    }

    const int nIdx  = lane & 15;
    const int mHalf = (lane >> 4) * 8;
    for (int mt = 0; mt < 2; ++mt) {
        for (int nt = 0; nt < 4; ++nt) {
            int c = n_base + nt*16 + nIdx;
            float bias = b_out[c];
            #pragma unroll
            for (int j = 0; j < 8; ++j) {
                int t = m_base + m0 + mt*16 + mHalf + j;   // token in rolled space
                int b = t / (HW*HW);
                int r = t - b*HW*HW;
                int i = r / HW, jj = r - i*HW;
                int i2 = (i + DISP) % HW, j2 = (jj + DISP) % HW;   // roll(+3,+3)
                out[((long)(b*HW + i2)*HW + j2)*EMBED + c] = acc[mt][nt][j] + bias;
            }
        }
    }
}

// ================= launcher =================
extern "C" void kernel_launch(void* const* d_in, const int* in_sizes, int n_in,
                              void* d_out, int out_size, void* d_ws, size_t ws_size,
                              hipStream_t stream) {
    (void)in_sizes; (void)n_in; (void)out_size; (void)ws_size;
    const float* x       = (const float*)d_in[0];
    const float* w_qkv   = (const float*)d_in[1];
    const float* pos_emb = (const float*)d_in[2];
    const float* w_out   = (const float*)d_in[3];
    const float* b_out   = (const float*)d_in[4];
    float* out = (float*)d_out;

    float* qkv     = (float*)d_ws;                         // 50176 x 1152
    float* attnout = qkv + (size_t)TOKENS * QKVC;          // 50176 x 384

    dim3 g1(TOKENS/BM, QKVC/BN);
    qkv_gemm<<<g1, 128, 0, stream>>>(x, w_qkv, qkv);

    attn_kernel<<<BATCH*HEADS*NWIN, 128, 0, stream>>>(qkv, pos_emb, attnout);

    dim3 g3(TOKENS/BM, EMBED/BN);
    out_gemm<<<g3, 128, 0, stream>>>(attnout, w_out, b_out, out);
}